// SimpleLNN_71193377898847
// MI455X (gfx1250) — compile-verified
//
#include <hip/hip_runtime.h>
#include <hip/hip_bf16.h>

// Fused Lagrangian-NN kernel for MI455X (gfx1250, wave32).
//   h  = relu(q @ W1^T + b1);  V = h @ W2^T + b2
//   L  = (q @ WL^T + bL).reshape(B,32,32);  M = L L^T
//   predicted_accel = M^{-1} action   (Cholesky, SPD)
// All GEMM work runs on V_WMMA_F32_16X16X4_F32 (native fp32 WMMA -> exact
// precision match with the fp32 reference). Fully fused: L (256MB) and h
// (64MB) never touch HBM.

typedef __attribute__((ext_vector_type(2))) float v2f;
typedef __attribute__((ext_vector_type(8))) float v8f;

#define B_TOTAL 65536
#define DDIM    32
#define HDIM    256
#define ROWS    16              // batch rows per workgroup (one WMMA tile)
#define LSTR    33              // padded row stride (odd -> conflict-free)
#define MATF    (32 * LSTR)     // floats per padded 32x32 matrix = 1056

// LDS layout (floats):
//   sQ : ROWS*LSTR            =   528
//   sL : ROWS*MATF            = 16896
//   sM : ROWS*MATF            = 16896
//   sV : ROWS                 =    16
#define SQ_OFF 0
#define SL_OFF 528
#define SM_OFF (528 + 16896)
#define SV_OFF (528 + 2 * 16896)
#define SMEM_FLOATS (528 + 2 * 16896 + 16)

__global__ void lnn_fused_kernel(const float* __restrict__ q,
                                 const float* __restrict__ action,
                                 const float* __restrict__ W1,
                                 const float* __restrict__ b1,
                                 const float* __restrict__ W2,
                                 const float* __restrict__ b2,
                                 const float* __restrict__ WL,
                                 const float* __restrict__ bL,
                                 float* __restrict__ out_pa,
                                 float* __restrict__ out_V) {
    extern __shared__ float smem[];
    float* sQ = smem + SQ_OFF;
    float* sL = smem + SL_OFF;
    float* sM = smem + SM_OFF;
    float* sV = smem + SV_OFF;

    const int tid  = threadIdx.x;           // 0..255 (8 wave32)
    const int lane = tid & 31;
    const int wave = tid >> 5;
    const int half = lane >> 4;             // 0 or 1 (lane half)
    const int mrow = lane & 15;             // row/col index within half
    const int row0 = blockIdx.x * ROWS;     // first batch row of this WG

    // ---- load q tile into LDS (16 x 32, padded stride 33) ----
    for (int i = tid; i < ROWS * DDIM; i += 256) {
        int r = i >> 5, c = i & 31;
        sQ[r * LSTR + c] = q[(size_t)(row0 + r) * DDIM + c];
    }
    if (tid < ROWS) sV[tid] = b2[0];
    __syncthreads();

    // =====================================================================
    // Stage A: WMMA GEMMs  D = q_tile(16x32) @ W^T_tile(32x16)
    //   tiles 0..15  -> W1 (h path, fold into V)
    //   tiles 16..79 -> WL (L path, store to sL)
    // =====================================================================
    for (int t = wave; t < 80; t += 8) {
        const bool  isH  = (t < 16);
        const int   col0 = isH ? (t << 4) : ((t - 16) << 4);
        const float* Wt  = isH ? W1 : WL;
        const float* bt  = isH ? b1 : bL;
        const int   myc  = col0 + mrow;          // this lane's output column
        const float* wrow = Wt + (size_t)myc * DDIM;

        v8f acc = {};
        #pragma unroll
        for (int kk = 0; kk < 8; ++kk) {
            const int k0 = (kk << 2) + (half << 1);   // K base per lane half
            v2f a, bf;
            a.x  = sQ[mrow * LSTR + k0];
            a.y  = sQ[mrow * LSTR + k0 + 1];
            bf.x = wrow[k0];
            bf.y = wrow[k0 + 1];
            acc = __builtin_amdgcn_wmma_f32_16x16x4_f32(
                      false, a, false, bf, (short)0, acc, false, false);
        }
        const float bias = bt[myc];
        if (isH) {
            const float w2 = W2[myc];
            #pragma unroll
            for (int r = 0; r < 8; ++r) {
                float h = fmaxf(acc[r] + bias, 0.0f);   // relu(q@W1^T + b1)
                float p = h * w2;                        // fold with W2
                p += __shfl_xor(p, 1, 32);
                p += __shfl_xor(p, 2, 32);
                p += __shfl_xor(p, 4, 32);
                p += __shfl_xor(p, 8, 32);               // sum over 16 cols
                if (mrow == 0) atomicAdd(&sV[r + (half << 3)], p); // ds_add_f32
            }
        } else {
            const int li = myc >> 5, lj = myc & 31;      // (i,j) inside 32x32 L
            #pragma unroll
            for (int r = 0; r < 8; ++r) {
                const int m = r + (half << 3);           // local batch row
                sL[m * MATF + li * LSTR + lj] = acc[r] + bias;
            }
        }
    }
    __syncthreads();

    if (tid < ROWS) out_V[row0 + tid] = sV[tid];

    // =====================================================================
    // Stage B: M = L L^T per batch via fp32 WMMA out of LDS.
    //   64 tiles: batch(16) x {I,J in {0,16}}^2
    //   A = L[I+m][k],  B[k][n] = L^T[k][J+n] = L[J+n][k]
    // =====================================================================
    for (int tt = wave; tt < 64; tt += 8) {
        const int bb  = tt >> 2;
        const int sub = tt & 3;
        const int I   = (sub >> 1) << 4;
        const int J   = (sub & 1) << 4;
        const float* Lb = sL + bb * MATF;

        v8f acc = {};
        #pragma unroll
        for (int kk = 0; kk < 8; ++kk) {
            const int k0 = (kk << 2) + (half << 1);
            v2f a, bf;
            a.x  = Lb[(I + mrow) * LSTR + k0];
            a.y  = Lb[(I + mrow) * LSTR + k0 + 1];
            bf.x = Lb[(J + mrow) * LSTR + k0];
            bf.y = Lb[(J + mrow) * LSTR + k0 + 1];
            acc = __builtin_amdgcn_wmma_f32_16x16x4_f32(
                      false, a, false, bf, (short)0, acc, false, false);
        }
        float* Mb = sM + bb * MATF;
        #pragma unroll
        for (int r = 0; r < 8; ++r)
            Mb[(I + r + (half << 3)) * LSTR + (J + mrow)] = acc[r];
    }
    __syncthreads();

    // =====================================================================
    // Stage C: per-batch Cholesky (M = Lc Lc^T) + two triangular solves.
    // One wave per batch, 2 passes. Lane i owns row i of M: every LDS
    // read-after-write inside the factorization is same-lane; cross-lane
    // values travel via shuffles. volatile pins LDS ordering (same-wave DS
    // ops are in-order per CDNA5 ISA).
    // =====================================================================
    for (int pass = 0; pass < 2; ++pass) {
        const int b = wave + (pass << 3);
        volatile float* Mb = sM + b * MATF;

        // ---- Cholesky, right-looking, lane-parallel over rows ----
        for (int k = 0; k < 32; ++k) {
            float colk = Mb[lane * LSTR + k];            // M[lane][k]
            float mkk  = __shfl(colk, k, 32);
            float dg   = sqrtf(mkk);
            float ljk  = (lane > k) ? (colk / dg)
                                    : ((lane == k) ? dg : colk);
            Mb[lane * LSTR + k] = ljk;                   // L[lane][k]
            for (int j = k + 1; j < 32; ++j) {
                float ljkj = __shfl(ljk, j, 32);         // L[j][k]
                if (lane >= j)
                    Mb[lane * LSTR + j] -= ljk * ljkj;   // trailing update
            }
        }

        // ---- forward solve  Lc y = action ----
        float rr = action[(size_t)(row0 + b) * DDIM + lane];
        for (int k = 0; k < 32; ++k) {
            float lkk = Mb[k * LSTR + k];
            float yk  = __shfl(rr, k, 32) / lkk;
            float lik = Mb[lane * LSTR + k];             // L[lane][k]
            if (lane > k)       rr -= lik * yk;
            else if (lane == k) rr = yk;
        }
        // ---- backward solve  Lc^T x = y ----
        for (int k = 31; k >= 0; --k) {
            float lkk = Mb[k * LSTR + k];
            float xk  = __shfl(rr, k, 32) / lkk;
            float lki = Mb[k * LSTR + lane];             // L[k][lane], lane<k
            if (lane < k)       rr -= lki * xk;
            else if (lane == k) rr = xk;
        }
        out_pa[(size_t)(row0 + b) * DDIM + lane] = rr;
    }
}

extern "C" void kernel_launch(void* const* d_in, const int* in_sizes, int n_in,
                              void* d_out, int out_size, void* d_ws, size_t ws_size,
                              hipStream_t stream) {
    (void)in_sizes; (void)n_in; (void)out_size; (void)d_ws; (void)ws_size;
    const float* q      = (const float*)d_in[0];
    // d_in[1] = q_dot : unused by the reference computation
    const float* action = (const float*)d_in[2];
    const float* W1     = (const float*)d_in[3];
    const float* b1     = (const float*)d_in[4];
    const float* W2     = (const float*)d_in[5];
    const float* b2     = (const float*)d_in[6];
    const float* WL     = (const float*)d_in[7];
    const float* bL     = (const float*)d_in[8];

    float* out_pa = (float*)d_out;                       // [B, 32]
    float* out_V  = out_pa + (size_t)B_TOTAL * DDIM;     // [B, 1]

    const size_t shmem = (size_t)SMEM_FLOATS * sizeof(float); // ~137 KB
    hipFuncSetAttribute(reinterpret_cast<const void*>(lnn_fused_kernel),
                        hipFuncAttributeMaxDynamicSharedMemorySize, (int)shmem);

    lnn_fused_kernel<<<dim3(B_TOTAL / ROWS), dim3(256), shmem, stream>>>(
        q, action, W1, b1, W2, b2, WL, bL, out_pa, out_V);
}